// KCN_GAT_50337016709818
// MI455X (gfx1250) — compile-verified
//
#include <hip/hip_runtime.h>
#include <hip/hip_bf16.h>
#include <math.h>

// ---------------------------------------------------------------------------
// Types for WMMA
// ---------------------------------------------------------------------------
typedef float v2f __attribute__((ext_vector_type(2)));
typedef float v8f __attribute__((ext_vector_type(8)));

#define D_FEAT 128

// ---------------------------------------------------------------------------
// Ordered-uint mapping for float atomic max (monotone, handles -inf)
// ---------------------------------------------------------------------------
__device__ __forceinline__ unsigned f2ord(float f) {
    unsigned b = __float_as_uint(f);
    return (b & 0x80000000u) ? ~b : (b | 0x80000000u);
}
__device__ __forceinline__ float ord2f(unsigned o) {
    unsigned b = (o & 0x80000000u) ? (o & 0x7fffffffu) : ~o;
    return __uint_as_float(b);
}

// ---------------------------------------------------------------------------
// Fill kernels
// ---------------------------------------------------------------------------
__global__ void k_fill_f32(float* p, float v, int n) {
    int i = blockIdx.x * blockDim.x + threadIdx.x;
    if (i < n) p[i] = v;
}
__global__ void k_fill_u32(unsigned* p, unsigned v, int n) {
    int i = blockIdx.x * blockDim.x + threadIdx.x;
    if (i < n) p[i] = v;
}

// ---------------------------------------------------------------------------
// GEMM: C[M,128] = A[M,128] @ B[128,128], fp32 via V_WMMA_F32_16X16X4_F32.
// One wave -> one 16x16 tile of C. 8 waves/block -> full 16-row strip.
// M must be a multiple of 16 (10000 = 625*16). EXEC all-ones throughout.
// ISA layouts (7.12.2): A 16x4: lane l<16 row l, v0=K0 v1=K1; lanes 16-31
// same rows, v0=K2 v1=K3. B 4x16 mirrors (row K striped over lanes).
// C/D: vgpr r, lanes0-15 -> M=r, lanes16-31 -> M=8+r, N=lane%16.
// ---------------------------------------------------------------------------
__global__ __launch_bounds__(256) void k_gemm_wmma_f32(
    const float* __restrict__ A, const float* __restrict__ B,
    float* __restrict__ C) {
    const int wave = threadIdx.x >> 5;          // 0..7 => n-tile
    const int lane = threadIdx.x & 31;
    const int g    = lane >> 4;                 // lane group 0/1
    const int l    = lane & 15;
    const int m0   = blockIdx.x * 16;
    const int n0   = wave * 16;

    const float* arow = A + (size_t)(m0 + l) * D_FEAT;
    v8f acc = {};
#pragma unroll 4
    for (int k = 0; k < D_FEAT; k += 4) {
        v2f a, b;
        a.x = arow[k + 2 * g];
        a.y = arow[k + 2 * g + 1];
        b.x = B[(size_t)(k + 2 * g) * D_FEAT + n0 + l];
        b.y = B[(size_t)(k + 2 * g + 1) * D_FEAT + n0 + l];
        acc = __builtin_amdgcn_wmma_f32_16x16x4_f32(
            /*neg_a=*/false, a, /*neg_b=*/false, b,
            /*c_mod=*/(short)0, acc, /*reuse_a=*/false, /*reuse_b=*/false);
    }
    float* ccol = C + n0 + l;
#pragma unroll
    for (int r = 0; r < 8; ++r)
        ccol[(size_t)(m0 + r + 8 * g) * D_FEAT] = acc[r];
}

// ---------------------------------------------------------------------------
// Per-node dot products: es[n] = h[n]·a_src ; ed[n] = h[n]·a_dst
// One wave per node, 4 columns per lane, wave-shuffle reduction (wave32).
// ---------------------------------------------------------------------------
__global__ __launch_bounds__(256) void k_node_dots(
    const float* __restrict__ h, const float* __restrict__ a_src,
    const float* __restrict__ a_dst, float* __restrict__ es,
    float* __restrict__ ed, int N) {
    int node = blockIdx.x * (blockDim.x >> 5) + (threadIdx.x >> 5);
    if (node >= N) return;
    int lane = threadIdx.x & 31;
    const float* row = h + (size_t)node * D_FEAT;
    float s = 0.f, d = 0.f;
#pragma unroll
    for (int c = lane; c < D_FEAT; c += 32) {
        float v = row[c];
        s += v * a_src[c];
        d += v * a_dst[c];
    }
#pragma unroll
    for (int off = 16; off > 0; off >>= 1) {
        s += __shfl_xor(s, off, 32);
        d += __shfl_xor(d, off, 32);
    }
    if (lane == 0) { es[node] = s; ed[node] = d; }
}

// ---------------------------------------------------------------------------
// Edge logits: e = leaky_relu(es[src]+ed[dst]); segment max into m_bits[dst]
// ---------------------------------------------------------------------------
__global__ void k_edge_logits(const long long* __restrict__ src,
                              const long long* __restrict__ dst,
                              const float* __restrict__ es,
                              const float* __restrict__ ed,
                              float* __restrict__ elog,
                              unsigned* __restrict__ m_bits, int E) {
    int i = blockIdx.x * blockDim.x + threadIdx.x;
    if (i >= E) return;
    int s = (int)src[i], d = (int)dst[i];
    float v = es[s] + ed[d];
    float e = v > 0.f ? v : 0.2f * v;
    elog[i] = e;
    atomicMax(&m_bits[d], f2ord(e));
}

// Decode segment max (clamp non-finite, matching reference semantics)
__global__ void k_decode_max(const unsigned* __restrict__ m_bits,
                             float* __restrict__ m, int N) {
    int i = blockIdx.x * blockDim.x + threadIdx.x;
    if (i >= N) return;
    float v = ord2f(m_bits[i]);
    m[i] = isfinite(v) ? v : 0.f;
}

// p = exp(e - m[dst]); segment sum into ssum[dst]
__global__ void k_exp_sum(const long long* __restrict__ dst,
                          const float* __restrict__ elog,
                          const float* __restrict__ m,
                          float* __restrict__ p, float* __restrict__ ssum,
                          int E) {
    int i = blockIdx.x * blockDim.x + threadIdx.x;
    if (i >= E) return;
    int d = (int)dst[i];
    float q = __expf(elog[i] - m[d]);
    p[i] = q;
    atomicAdd(&ssum[d], q);
}

// ---------------------------------------------------------------------------
// Scatter: acc[dst] += h[src] * (p / (ssum[dst]+1e-16)). One wave per edge,
// float4 gather per lane (128 floats = 32 lanes x float4), f32 atomic adds.
// ---------------------------------------------------------------------------
__global__ __launch_bounds__(256) void k_edge_scatter(
    const long long* __restrict__ src, const long long* __restrict__ dst,
    const float* __restrict__ h, const float* __restrict__ p,
    const float* __restrict__ ssum, float* __restrict__ acc, int E) {
    int e = blockIdx.x * (blockDim.x >> 5) + (threadIdx.x >> 5);
    if (e >= E) return;
    int lane = threadIdx.x & 31;
    int s = (int)src[e], d = (int)dst[e];
    float alpha = p[e] / (ssum[d] + 1e-16f);
    const float4* hr = (const float4*)(h + (size_t)s * D_FEAT);
    float* ar = acc + (size_t)d * D_FEAT;
    float4 v = hr[lane];
    atomicAdd(&ar[lane * 4 + 0], v.x * alpha);
    atomicAdd(&ar[lane * 4 + 1], v.y * alpha);
    atomicAdd(&ar[lane * 4 + 2], v.z * alpha);
    atomicAdd(&ar[lane * 4 + 3], v.w * alpha);
}

// out = relu(acc + bias)
__global__ void k_bias_relu(const float* __restrict__ acc,
                            const float* __restrict__ bias,
                            float* __restrict__ out, int total) {
    int i = blockIdx.x * blockDim.x + threadIdx.x;
    if (i >= total) return;
    float v = acc[i] + bias[i & (D_FEAT - 1)];
    out[i] = v > 0.f ? v : 0.f;
}

// ---------------------------------------------------------------------------
// Host-side launch
// ---------------------------------------------------------------------------
extern "C" void kernel_launch(void* const* d_in, const int* in_sizes, int n_in,
                              void* d_out, int out_size, void* d_ws,
                              size_t ws_size, hipStream_t stream) {
    const float*     x      = (const float*)d_in[0];
    const long long* eidx   = (const long long*)d_in[1];  // int64 [2,E]
    // d_in[2] = edge_weight: unused by the reference math
    const float* W0 = (const float*)d_in[3];
    const float* as0 = (const float*)d_in[4];
    const float* ad0 = (const float*)d_in[5];
    const float* b0 = (const float*)d_in[6];
    const float* W1 = (const float*)d_in[7];
    const float* as1 = (const float*)d_in[8];
    const float* ad1 = (const float*)d_in[9];
    const float* b1 = (const float*)d_in[10];

    const int N = in_sizes[0] / D_FEAT;   // 10000
    const int E = in_sizes[1] / 2;        // 640000
    const long long* src = eidx;
    const long long* dst = eidx + E;
    const size_t ND = (size_t)N * D_FEAT;

    // Carve workspace (256B aligned regions)
    char* base = (char*)d_ws;
    auto carve = [&](size_t bytes) {
        char* p = base;
        base += (bytes + 255) & ~(size_t)255;
        return p;
    };
    float*    h      = (float*)carve(ND * 4);
    float*    acc    = (float*)carve(ND * 4);
    float*    x1     = (float*)carve(ND * 4);
    float*    es     = (float*)carve((size_t)N * 4);
    float*    ed     = (float*)carve((size_t)N * 4);
    unsigned* m_bits = (unsigned*)carve((size_t)N * 4);
    float*    m      = (float*)carve((size_t)N * 4);
    float*    ssum   = (float*)carve((size_t)N * 4);
    float*    elog   = (float*)carve((size_t)E * 4);
    float*    p      = (float*)carve((size_t)E * 4);

    const int TB = 256;
    const dim3 blk(TB);
    const int gN    = (N + TB - 1) / TB;
    const int gND   = (int)((ND + TB - 1) / TB);
    const int gE    = (E + TB - 1) / TB;
    const int gNw   = (N + 7) / 8;   // wave-per-node kernels (8 waves/block)
    const int gEw   = (E + 7) / 8;   // wave-per-edge kernels
    const int gGemm = N / 16;        // 625 blocks, 8 WMMA waves each

    auto layer = [&](const float* xin, const float* W, const float* a_s,
                     const float* a_d, const float* bias, float* out) {
        // h = xin @ W  (WMMA fp32)
        k_gemm_wmma_f32<<<gGemm, blk, 0, stream>>>(xin, W, h);
        // es/ed per node
        k_node_dots<<<gNw, blk, 0, stream>>>(h, a_s, a_d, es, ed, N);
        // init segment-max bits to ord(-inf) = ~bits(-inf) = 0x007FFFFF
        k_fill_u32<<<gN, blk, 0, stream>>>(m_bits, 0x007FFFFFu, N);
        k_fill_f32<<<gN, blk, 0, stream>>>(ssum, 0.f, N);
        k_fill_f32<<<gND, blk, 0, stream>>>(acc, 0.f, (int)ND);
        // edge logits + segment max
        k_edge_logits<<<gE, blk, 0, stream>>>(src, dst, es, ed, elog, m_bits, E);
        k_decode_max<<<gN, blk, 0, stream>>>(m_bits, m, N);
        // softmax numerator + segment sum
        k_exp_sum<<<gE, blk, 0, stream>>>(dst, elog, m, p, ssum, E);
        // weighted scatter
        k_edge_scatter<<<gEw, blk, 0, stream>>>(src, dst, h, p, ssum, acc, E);
        // bias + relu
        k_bias_relu<<<gND, blk, 0, stream>>>(acc, bias, out, (int)ND);
    };

    (void)n_in; (void)out_size; (void)ws_size;

    layer(x,  W0, as0, ad0, b0, x1);
    layer(x1, W1, as1, ad1, b1, (float*)d_out);
}